// SelentionLinear_62878321214026
// MI455X (gfx1250) — compile-verified
//
#include <hip/hip_runtime.h>
#include <hip/hip_bf16.h>

#define NPTS   56564
#define BATCH  8
#define KMEANS 8
#define HDIM   256
#define SPLIT  32

typedef __attribute__((ext_vector_type(2))) float v2f;
typedef __attribute__((ext_vector_type(8))) float v8f;

// ---------------------------------------------------------------------------
// Kernel 1: per (batch, split-chunk) partial argmin of squared distance to all
// 8 means. Reads pos exactly once (the only mandatory HBM traffic, ~5.4 MB).
// ---------------------------------------------------------------------------
__global__ void sel_argmin_partial(const float* __restrict__ pos,
                                   const float* __restrict__ mean,
                                   float* __restrict__ pv,
                                   unsigned* __restrict__ pi) {
  const int b   = blockIdx.x / SPLIT;
  const int s   = blockIdx.x % SPLIT;
  const int tid = threadIdx.x;
  const int chunk = (NPTS + SPLIT - 1) / SPLIT;
  const int n0 = s * chunk;
  const int n1 = (n0 + chunk < NPTS) ? (n0 + chunk) : NPTS;

  float mx[KMEANS], my[KMEANS], mz[KMEANS];
#pragma unroll
  for (int k = 0; k < KMEANS; ++k) {
    mx[k] = mean[k * 3 + 0];
    my[k] = mean[k * 3 + 1];
    mz[k] = mean[k * 3 + 2];
  }

  float    bestv[KMEANS];
  unsigned besti[KMEANS];
#pragma unroll
  for (int k = 0; k < KMEANS; ++k) { bestv[k] = 3.4e38f; besti[k] = 0xFFFFFFFFu; }

  const float* pb = pos + (size_t)b * NPTS * 3;
  for (int n = n0 + tid; n < n1; n += 256) {
    const float px = pb[n * 3 + 0];
    const float py = pb[n * 3 + 1];
    const float pz = pb[n * 3 + 2];
#pragma unroll
    for (int k = 0; k < KMEANS; ++k) {
      const float dx = px - mx[k];
      const float dy = py - my[k];
      const float dz = pz - mz[k];
      const float d2 = dx * dx + dy * dy + dz * dz;
      // strictly-less keeps first occurrence on exact ties (matches argmax)
      if (d2 < bestv[k]) { bestv[k] = d2; besti[k] = (unsigned)n; }
    }
  }

  __shared__ float    sv[KMEANS][256];
  __shared__ unsigned si[KMEANS][256];
#pragma unroll
  for (int k = 0; k < KMEANS; ++k) { sv[k][tid] = bestv[k]; si[k][tid] = besti[k]; }
  __syncthreads();

  for (int stride = 128; stride > 0; stride >>= 1) {
    if (tid < stride) {
#pragma unroll
      for (int k = 0; k < KMEANS; ++k) {
        const float    va = sv[k][tid];
        const unsigned ia = si[k][tid];
        const float    vb = sv[k][tid + stride];
        const unsigned ib = si[k][tid + stride];
        const bool take_b = (vb < va) || (vb == va && ib < ia);
        sv[k][tid] = take_b ? vb : va;
        si[k][tid] = take_b ? ib : ia;
      }
    }
    __syncthreads();
  }

  if (tid == 0) {
#pragma unroll
    for (int k = 0; k < KMEANS; ++k) {
      pv[((size_t)b * SPLIT + s) * KMEANS + k] = sv[k][0];
      pi[((size_t)b * SPLIT + s) * KMEANS + k] = si[k][0];
    }
  }
}

// ---------------------------------------------------------------------------
// Kernel 2: final reduce over splits, per-batch dedupe (= the clip/clamp),
// gather x values. Emits A diag values aval[64] and selected rows nsel[64].
// ---------------------------------------------------------------------------
__global__ void sel_finalize(const float* __restrict__ x,
                             const float* __restrict__ pv,
                             const unsigned* __restrict__ pi,
                             float* __restrict__ aval,
                             unsigned* __restrict__ nsel) {
  const int t = threadIdx.x;           // 64 threads: t = b*8 + k
  const int b = t >> 3;
  const int k = t & 7;

  float    bv = 3.4e38f;
  unsigned bi = 0xFFFFFFFFu;
  for (int s = 0; s < SPLIT; ++s) {
    const float    v = pv[((size_t)b * SPLIT + s) * KMEANS + k];
    const unsigned i = pi[((size_t)b * SPLIT + s) * KMEANS + k];
    const bool take = (v < bv) || (v == bv && i < bi);
    if (take) { bv = v; bi = i; }
  }

  __shared__ unsigned nb[BATCH * KMEANS];
  nb[t] = bi;
  __syncthreads();

  bool dup = false;
  for (int kk = 0; kk < k; ++kk)
    if (nb[b * KMEANS + kk] == bi) dup = true;

  nsel[t] = bi;
  aval[t] = dup ? 0.0f : x[(size_t)b * NPTS + bi];
}

// ---------------------------------------------------------------------------
// Kernel 3: out[16(pad8) x 256] = A[16 x 64] * Wsel[64 x 256] + bias via
// v_wmma_f32_16x16x4_f32 (exact fp32). One wave per 16-wide H tile.
// A is block diagonal: A[m, j] = (j>>3 == m) ? aval[j] : 0.
// fp32 16x4 A layout: lanes 0-15 = M, VGPR0/1 = K pair, half-wave picks K+2.
// ---------------------------------------------------------------------------
__global__ void sel_wmma_gemm(const float* __restrict__ weight,
                              const float* __restrict__ bias,
                              const float* __restrict__ aval,
                              const unsigned* __restrict__ nsel,
                              float* __restrict__ out) {
  const int lane = threadIdx.x;        // 32 lanes, one wave
  const int tile = blockIdx.x;         // 0..15 -> columns [tile*16, tile*16+16)
  const int col  = tile * 16 + (lane & 15);
  const int half = lane >> 4;          // 0: K={k0,k0+1}, 1: K={k0+2,k0+3}
  const int rowA = lane & 15;          // M index for A operand

  v8f acc = {};
#pragma unroll
  for (int k0 = 0; k0 < BATCH * KMEANS; k0 += 4) {
    const int j0 = k0 + half * 2;
    const int j1 = j0 + 1;

    v2f a;
    a.x = ((j0 >> 3) == rowA) ? aval[j0] : 0.0f;
    a.y = ((j1 >> 3) == rowA) ? aval[j1] : 0.0f;

    v2f bm;
    bm.x = weight[(size_t)nsel[j0] * HDIM + col];
    bm.y = weight[(size_t)nsel[j1] * HDIM + col];

    acc = __builtin_amdgcn_wmma_f32_16x16x4_f32(
        /*neg_a=*/false, a, /*neg_b=*/false, bm,
        /*c_mod=*/(short)0, acc, /*reuse_a=*/false, /*reuse_b=*/false);
  }

  if (half == 0) {                     // D rows 0..7 (lanes 0-15) are batches
    const float bv = bias[col];
#pragma unroll
    for (int i = 0; i < BATCH; ++i)
      out[i * HDIM + col] = acc[i] + bv;
  }
}

extern "C" void kernel_launch(void* const* d_in, const int* in_sizes, int n_in,
                              void* d_out, int out_size, void* d_ws, size_t ws_size,
                              hipStream_t stream) {
  const float* x      = (const float*)d_in[0];  // [8, 56564]
  const float* pos    = (const float*)d_in[1];  // [8, 56564, 3]
  const float* mean   = (const float*)d_in[2];  // [8, 3]
  const float* weight = (const float*)d_in[3];  // [1, 56564, 256]
  const float* bias   = (const float*)d_in[4];  // [1, 256]
  float* out = (float*)d_out;                   // [8, 256]

  // workspace layout (all 4B aligned, ~17 KB total)
  char* ws = (char*)d_ws;
  float*    pv   = (float*)(ws + 0);                       // 64*32 floats = 8 KB
  unsigned* pi   = (unsigned*)(ws + 8192);                 // 8 KB
  float*    aval = (float*)(ws + 16384);                   // 256 B
  unsigned* nsel = (unsigned*)(ws + 16384 + 256);          // 256 B

  sel_argmin_partial<<<BATCH * SPLIT, 256, 0, stream>>>(pos, mean, pv, pi);
  sel_finalize<<<1, BATCH * KMEANS, 0, stream>>>(x, pv, pi, aval, nsel);
  sel_wmma_gemm<<<HDIM / 16, 32, 0, stream>>>(weight, bias, aval, nsel, out);
}